// HeteroRGCN_86861418595103
// MI455X (gfx1250) — compile-verified
//
#include <hip/hip_runtime.h>

typedef __attribute__((ext_vector_type(2))) float v2f;
typedef __attribute__((ext_vector_type(8))) float v8f;

#define N_NODES 50000
#define N_EDGES 800000
#define IN_F    128
#define HID_F   128
#define OUT_F   64

// ---------------------------------------------------------------------------
// utility kernels
// ---------------------------------------------------------------------------
__global__ void zero_kernel(float* __restrict__ p, long long n) {
    long long i = (long long)blockIdx.x * blockDim.x + threadIdx.x;
    if (i < n) p[i] = 0.0f;
}

__global__ void leaky_relu_kernel(float* __restrict__ p, long long n) {
    long long i = (long long)blockIdx.x * blockDim.x + threadIdx.x;
    if (i < n) {
        float v = p[i];
        p[i] = v > 0.0f ? v : 0.01f * v;
    }
}

// ---------------------------------------------------------------------------
// fp32 GEMM via V_WMMA_F32_16X16X4_F32.
//   C[M,N] = A[M,K] @ W[K,N] + bias[N]
// One wave owns a 16-row stripe across all N (NT = N/16 tiles of 16 cols),
// reusing the A fragment across n-tiles. M must be a multiple of 16.
//
// Fragment layouts per CDNA5 ISA 7.12.2 (32-bit):
//   A 16x4 : lane&15 = M row; lane>>4 selects K pair {0,1} vs {2,3};
//            v0 = K even, v1 = K odd of the pair.
//   B 4x16 : lane&15 = N col; lane>>4 selects K pair; v0/v1 = the two rows.
//   C 16x16: vgpr i, lanes 0-15 -> M=i, lanes 16-31 -> M=i+8; lane&15 = N col.
// ---------------------------------------------------------------------------
template<int NT>
__global__ __launch_bounds__(256)
void gemm_wmma_kernel(const float* __restrict__ A,
                      const float* __restrict__ W,
                      const float* __restrict__ bias,
                      float* __restrict__ C,
                      int M, int K) {
    const int N    = NT * 16;
    const int lane = threadIdx.x & 31;
    const int wave = threadIdx.x >> 5;
    const int m0   = (blockIdx.x * 8 + wave) * 16;   // uniform per wave
    if (m0 >= M) return;                              // whole-wave exit (EXEC stays all-1)

    const int n  = lane & 15;        // column within 16-wide tile / row of A frag
    const int kh = (lane >> 4) * 2;  // K-pair select: 0 or 2

    v8f acc[NT];
#pragma unroll
    for (int t = 0; t < NT; ++t) {
        float bv = bias[t * 16 + n];
#pragma unroll
        for (int i = 0; i < 8; ++i) acc[t][i] = bv;
    }

    for (int k0 = 0; k0 < K; k0 += 4) {
        const float* arow = A + (size_t)(m0 + n) * K + (k0 + kh);
        v2f a;
        a.x = arow[0];
        a.y = arow[1];
#pragma unroll
        for (int t = 0; t < NT; ++t) {
            const float* bcol = W + (size_t)(k0 + kh) * N + t * 16 + n;
            v2f b;
            b.x = bcol[0];
            b.y = bcol[N];
            acc[t] = __builtin_amdgcn_wmma_f32_16x16x4_f32(
                false, a, false, b, (short)0, acc[t], false, false);
        }
    }

    const int rbase = m0 + (lane >> 4) * 8;
#pragma unroll
    for (int t = 0; t < NT; ++t) {
#pragma unroll
        for (int i = 0; i < 8; ++i) {
            C[(size_t)(rbase + i) * N + t * 16 + n] = acc[t][i];
        }
    }
}

// ---------------------------------------------------------------------------
// Edge scatter: acc[dst] += Wh[src] (per feature), deg[dst] += 1.
// Each thread handles one float4 chunk of one edge's message.
// ---------------------------------------------------------------------------
template<int F>
__global__ void scatter_kernel(const float* __restrict__ Wh,
                               const int* __restrict__ src,
                               const int* __restrict__ dst,
                               float* __restrict__ acc,
                               float* __restrict__ deg,
                               int n_edges) {
    const int CH = F / 4;
    long long gid = (long long)blockIdx.x * blockDim.x + threadIdx.x;
    if (gid >= (long long)n_edges * CH) return;
    int e = (int)(gid / CH);
    int c = (int)(gid - (long long)e * CH);
    int s = src[e];
    int d = dst[e];
    float4 m = *(const float4*)(Wh + (size_t)s * F + c * 4);
    float* a = acc + (size_t)d * F + c * 4;
    atomicAdd(a + 0, m.x);
    atomicAdd(a + 1, m.y);
    atomicAdd(a + 2, m.z);
    atomicAdd(a + 3, m.w);
    if (c == 0) atomicAdd(deg + d, 1.0f);
}

// ---------------------------------------------------------------------------
// h += acc / max(deg, 1)
// ---------------------------------------------------------------------------
template<int F>
__global__ void combine_kernel(const float* __restrict__ acc,
                               const float* __restrict__ deg,
                               float* __restrict__ h,
                               long long n) {
    long long i = (long long)blockIdx.x * blockDim.x + threadIdx.x;
    if (i >= n) return;
    float dg = fmaxf(deg[i / F], 1.0f);
    h[i] += acc[i] / dg;
}

// ---------------------------------------------------------------------------
// launcher
// ---------------------------------------------------------------------------
extern "C" void kernel_launch(void* const* d_in, const int* in_sizes, int n_in,
                              void* d_out, int out_size, void* d_ws, size_t ws_size,
                              hipStream_t stream) {
    const float* feat = (const float*)d_in[0];
    const float* W1   = (const float*)d_in[1];
    const float* b1   = (const float*)d_in[2];
    const float* W2   = (const float*)d_in[3];
    const float* b2   = (const float*)d_in[4];
    const int*   es1  = (const int*)d_in[5];
    const int*   ed1  = (const int*)d_in[6];
    const int*   es2  = (const int*)d_in[7];
    const int*   ed2  = (const int*)d_in[8];
    float* out = (float*)d_out;

    float* Wh  = (float*)d_ws;                      // 50000*128 f32
    float* acc = Wh  + (size_t)N_NODES * HID_F;     // 50000*128 f32
    float* h1  = acc + (size_t)N_NODES * HID_F;     // 50000*128 f32
    float* deg = h1  + (size_t)N_NODES * HID_F;     // 50000   f32

    const long long nh1  = (long long)N_NODES * HID_F;
    const long long nout = (long long)N_NODES * OUT_F;
    const int gemm_blocks = (N_NODES / 16 + 7) / 8;  // 8 waves (m-tiles) per block

    auto zero = [&](float* p, long long n) {
        zero_kernel<<<(unsigned)((n + 255) / 256), 256, 0, stream>>>(p, n);
    };

    // -------- layer 1: h1 = sum_r mean_dst( (feat @ W1[r] + b1[r])[src] ) ----
    zero(h1, nh1);
    for (int r = 0; r < 3; ++r) {
        gemm_wmma_kernel<HID_F / 16><<<gemm_blocks, 256, 0, stream>>>(
            feat, W1 + (size_t)r * IN_F * HID_F, b1 + (size_t)r * HID_F,
            Wh, N_NODES, IN_F);
        zero(acc, nh1);
        zero(deg, N_NODES);
        long long work = (long long)N_EDGES * (HID_F / 4);
        scatter_kernel<HID_F><<<(unsigned)((work + 255) / 256), 256, 0, stream>>>(
            Wh, es1 + (size_t)r * N_EDGES, ed1 + (size_t)r * N_EDGES,
            acc, deg, N_EDGES);
        combine_kernel<HID_F><<<(unsigned)((nh1 + 255) / 256), 256, 0, stream>>>(
            acc, deg, h1, nh1);
    }
    leaky_relu_kernel<<<(unsigned)((nh1 + 255) / 256), 256, 0, stream>>>(h1, nh1);

    // -------- layer 2: out = sum_r mean_dst( (h1 @ W2[r] + b2[r])[src] ) -----
    zero(out, nout);
    for (int r = 0; r < 3; ++r) {
        gemm_wmma_kernel<OUT_F / 16><<<gemm_blocks, 256, 0, stream>>>(
            h1, W2 + (size_t)r * HID_F * OUT_F, b2 + (size_t)r * OUT_F,
            Wh, N_NODES, HID_F);
        zero(acc, nout);
        zero(deg, N_NODES);
        long long work = (long long)N_EDGES * (OUT_F / 4);
        scatter_kernel<OUT_F><<<(unsigned)((work + 255) / 256), 256, 0, stream>>>(
            Wh, es2 + (size_t)r * N_EDGES, ed2 + (size_t)r * N_EDGES,
            acc, deg, N_EDGES);
        combine_kernel<OUT_F><<<(unsigned)((nout + 255) / 256), 256, 0, stream>>>(
            acc, deg, out, nout);
    }
}